// ProposalLayer_91087666413893
// MI455X (gfx1250) — compile-verified
//
#include <hip/hip_runtime.h>
#include <hip/hip_bf16.h>
#include <stdint.h>

// Problem constants (match reference)
#define B       16
#define A       262144
#define TOP_N   64
#define IOU_THR 0.7f

// Tuning
#define NBINS   2048    // score histogram bins over sigmoid output (0,1)
#define TARGET  1536    // aim for ~top-1536 candidates per batch
#define CAP     2048    // candidate buffer capacity per batch (fits LDS)
#define THREADS 256
#define ITEMS   16
#define ITEMS4  (ITEMS / 4)             // float4 iterations per thread
#define BLOCK_ELEMS (THREADS * ITEMS)   // 4096 elems per block
#define GRID_X  (A / BLOCK_ELEMS)       // 64 blocks per batch in scan passes
#define NWAVES  (THREADS / 32)          // wave32

__device__ __forceinline__ float sigmoidf_fast(float x) {
    return 1.0f / (1.0f + __expf(-x));
}
__device__ __forceinline__ float clamp01(float v) {
    return fminf(fmaxf(v, 0.0f), 1.0f);
}
__device__ __forceinline__ int score_bin(float s) {
    int bin = (int)(s * (float)NBINS);
    return bin < 0 ? 0 : (bin > NBINS - 1 ? NBINS - 1 : bin);
}
// order-preserving float -> uint32 (total order matches float compare)
__device__ __forceinline__ uint32_t ord32(float f) {
    uint32_t u = __float_as_uint(f);
    return (u & 0x80000000u) ? ~u : (u | 0x80000000u);
}

// ---------------------------------------------------------------- pass 0: zero
__global__ void zero_hist_kernel(int* __restrict__ hist) {
    int i = blockIdx.x * blockDim.x + threadIdx.x;
    if (i < B * NBINS) hist[i] = 0;
}

// ------------------------------------------------------ pass 1: score histogram
__global__ void __launch_bounds__(THREADS)
hist_kernel(const float* __restrict__ cls, int* __restrict__ hist) {
    __shared__ int sh[NBINS];
    const int b = blockIdx.y;
    for (int i = threadIdx.x; i < NBINS; i += THREADS) sh[i] = 0;
    __syncthreads();
    const float4* __restrict__ cls4 = (const float4*)cls;
    const size_t base4 = ((size_t)b * A + (size_t)blockIdx.x * BLOCK_ELEMS) / 4;
    for (int k = 0; k < ITEMS4; ++k) {
        const size_t vi = base4 + threadIdx.x + (size_t)k * THREADS;
        __builtin_prefetch(cls4 + vi + 2 * THREADS, 0, 0);  // global_prefetch_b8
        const float4 c = cls4[vi];
        atomicAdd(&sh[score_bin(sigmoidf_fast(c.x))], 1);
        atomicAdd(&sh[score_bin(sigmoidf_fast(c.y))], 1);
        atomicAdd(&sh[score_bin(sigmoidf_fast(c.z))], 1);
        atomicAdd(&sh[score_bin(sigmoidf_fast(c.w))], 1);
    }
    __syncthreads();
    for (int i = threadIdx.x; i < NBINS; i += THREADS) {
        const int c = sh[i];
        if (c) atomicAdd(&hist[b * NBINS + i], c);
    }
}

// --------------------------------------------- pass 2: per-batch threshold bin
__global__ void thresh_kernel(const int* __restrict__ hist,
                              int* __restrict__ tbin,
                              int* __restrict__ candCount) {
    const int b = threadIdx.x;
    if (b >= B) return;
    int cum = 0, t = 0;
    for (int bin = NBINS - 1; bin >= 0; --bin) {
        cum += hist[b * NBINS + bin];
        if (cum >= TARGET) { t = bin; break; }
    }
    tbin[b] = t;
    candCount[b] = 0;
}

// ------------------------- pass 3: compact survivors, decode+clip their boxes
__global__ void __launch_bounds__(THREADS)
compact_kernel(const float* __restrict__ cls,
               const float* __restrict__ reg,
               const float* __restrict__ anchors,
               const int* __restrict__ tbin,
               int* __restrict__ candCount,
               float* __restrict__ candScore,
               float* __restrict__ candBox) {
    const int b = blockIdx.y;
    const int t = tbin[b];
    const float4* __restrict__ cls4 = (const float4*)cls;
    const float4* __restrict__ reg4 = (const float4*)reg;        // one float4 per anchor
    const float4* __restrict__ anc4 = (const float4*)anchors;    // one float4 per anchor
    float4* __restrict__ cbox4 = (float4*)candBox;
    const size_t base4 = ((size_t)b * A + (size_t)blockIdx.x * BLOCK_ELEMS) / 4;
    for (int k = 0; k < ITEMS4; ++k) {
        const size_t vi = base4 + threadIdx.x + (size_t)k * THREADS;
        __builtin_prefetch(cls4 + vi + 2 * THREADS, 0, 0);
        const float4 c = cls4[vi];
        const float sv[4] = { sigmoidf_fast(c.x), sigmoidf_fast(c.y),
                              sigmoidf_fast(c.z), sigmoidf_fast(c.w) };
        #pragma unroll
        for (int j = 0; j < 4; ++j) {
            if (score_bin(sv[j]) >= t) {
                const int pos = atomicAdd(&candCount[b], 1);
                if (pos < CAP) {
                    const size_t gi = vi * 4 + j;              // global anchor slot
                    const size_t ai = gi - (size_t)b * A;      // anchor index
                    const float4 a = anc4[ai];
                    const float4 d = reg4[gi];
                    const float ah = a.z - a.x, aw = a.w - a.y;
                    const float acy = a.x + 0.5f * ah, acx = a.y + 0.5f * aw;
                    const float cy = d.x * ah + acy;
                    const float cx = d.y * aw + acx;
                    const float h  = __expf(d.z) * ah;
                    const float w  = __expf(d.w) * aw;
                    float4 obx;
                    obx.x = clamp01(cy - 0.5f * h);
                    obx.y = clamp01(cx - 0.5f * w);
                    obx.z = clamp01(cy + 0.5f * h);
                    obx.w = clamp01(cx + 0.5f * w);
                    cbox4[(size_t)b * CAP + pos] = obx;
                    candScore[(size_t)b * CAP + pos] = sv[j];
                }
            }
        }
    }
}

// ------------------------------ pass 4: per-batch in-LDS NMS (one WG per batch)
__global__ void __launch_bounds__(THREADS)
nms_kernel(const int* __restrict__ candCount,
           const float* __restrict__ candScore,
           const float* __restrict__ candBox,
           float* __restrict__ out) {
    __shared__ float sScore[CAP];
    __shared__ float sArea[CAP];
    __shared__ float sBox[CAP * 4];
    __shared__ unsigned long long redK[NWAVES];
    __shared__ float sel[4];

    const int b = blockIdx.x;
    int n = candCount[b];
    if (n > CAP) n = CAP;

    // indices output (reference returns all-zero int32; 0.0f is bitwise 0)
    if (threadIdx.x < TOP_N)
        out[B * TOP_N * 4 + b * TOP_N + threadIdx.x] = 0.0f;

    // CDNA5 async global->LDS staging of candidates (ASYNCcnt path)
    const float* gScore = candScore + (size_t)b * CAP;
    const float* gBox   = candBox   + (size_t)b * CAP * 4;
    for (int i = threadIdx.x; i < n; i += THREADS) {
        const uint32_t lds_box = (uint32_t)(uintptr_t)(&sBox[i * 4]);
        const uint64_t g_box   = (uint64_t)(uintptr_t)(gBox + (size_t)i * 4);
        asm volatile("global_load_async_to_lds_b128 %0, %1, off"
                     :: "v"(lds_box), "v"(g_box) : "memory");
        const uint32_t lds_sc = (uint32_t)(uintptr_t)(&sScore[i]);
        const uint64_t g_sc   = (uint64_t)(uintptr_t)(gScore + i);
        asm volatile("global_load_async_to_lds_b32 %0, %1, off"
                     :: "v"(lds_sc), "v"(g_sc) : "memory");
    }
    asm volatile("s_wait_asynccnt 0" ::: "memory");
    __syncthreads();

    // pad tail + precompute areas
    for (int i = threadIdx.x; i < CAP; i += THREADS) {
        if (i >= n) {
            sScore[i] = -2.0f;
            sArea[i]  = 0.0f;
        } else {
            const float y1 = sBox[i * 4 + 0], x1 = sBox[i * 4 + 1];
            const float y2 = sBox[i * 4 + 2], x2 = sBox[i * 4 + 3];
            sArea[i] = fmaxf(y2 - y1, 0.0f) * fmaxf(x2 - x1, 0.0f);
        }
    }
    __syncthreads();

    const int wid  = threadIdx.x >> 5;
    const int lane = threadIdx.x & 31;

    for (int it = 0; it < TOP_N; ++it) {
        // strided local argmax with packed key: [ord(score) | (0xFFFFFFFF - idx)]
        unsigned long long key = 0ull;
        for (int i = threadIdx.x; i < CAP; i += THREADS) {
            const unsigned long long k =
                ((unsigned long long)ord32(sScore[i]) << 32) |
                (unsigned long long)(0xFFFFFFFFu - (uint32_t)i);
            if (k > key) key = k;
        }
        // wave32 shuffle reduction (no barriers)
        #pragma unroll
        for (int m = 16; m > 0; m >>= 1) {
            const unsigned long long o = __shfl_xor((unsigned long long)key, m, 32);
            if (o > key) key = o;
        }
        if (lane == 0) redK[wid] = key;
        __syncthreads();
        if (threadIdx.x == 0) {
            unsigned long long bk = redK[0];
            #pragma unroll
            for (int w = 1; w < NWAVES; ++w)
                if (redK[w] > bk) bk = redK[w];
            const int idx = (int)(0xFFFFFFFFu - (uint32_t)bk);
            const float sc = sScore[idx];
            const float v = (sc > -0.5f) ? 1.0f : 0.0f;
            sel[0] = sBox[idx * 4 + 0];
            sel[1] = sBox[idx * 4 + 1];
            sel[2] = sBox[idx * 4 + 2];
            sel[3] = sBox[idx * 4 + 3];
            float* orow = &out[((size_t)b * TOP_N + it) * 4];
            orow[0] = sel[0] * v;
            orow[1] = sel[1] * v;
            orow[2] = sel[2] * v;
            orow[3] = sel[3] * v;
            sScore[idx] = -1.0f;
        }
        __syncthreads();

        // suppression sweep
        const float by1 = sel[0], bx1 = sel[1], by2 = sel[2], bx2 = sel[3];
        const float bArea = fmaxf(by2 - by1, 0.0f) * fmaxf(bx2 - bx1, 0.0f);
        for (int i = threadIdx.x; i < n; i += THREADS) {
            const float iy1 = fmaxf(sBox[i * 4 + 0], by1);
            const float ix1 = fmaxf(sBox[i * 4 + 1], bx1);
            const float iy2 = fminf(sBox[i * 4 + 2], by2);
            const float ix2 = fminf(sBox[i * 4 + 3], bx2);
            const float inter = fmaxf(iy2 - iy1, 0.0f) * fmaxf(ix2 - ix1, 0.0f);
            const float uni = sArea[i] + bArea - inter;
            const float iou = (uni > 0.0f) ? (inter / uni) : 0.0f;
            if (iou > IOU_THR) sScore[i] = -1.0f;
        }
        __syncthreads();
    }
}

// ---------------------------------------------------------------------- launch
extern "C" void kernel_launch(void* const* d_in, const int* in_sizes, int n_in,
                              void* d_out, int out_size, void* d_ws, size_t ws_size,
                              hipStream_t stream) {
    const float* cls     = (const float*)d_in[0];   // rpn_cls  (B,A,1)
    const float* reg     = (const float*)d_in[1];   // rpn_reg  (B,A,4)
    const float* anchors = (const float*)d_in[2];   // anchors  (A,4)
    float* out = (float*)d_out;                     // proposals (B*TOP_N,4) ++ indices (B*TOP_N)

    char* ws = (char*)d_ws;
    int* hist      = (int*)ws;               ws += (size_t)B * NBINS * sizeof(int);
    int* tbin      = (int*)ws;               ws += (size_t)B * sizeof(int);
    int* candCount = (int*)ws;               ws += (size_t)B * sizeof(int);
    uintptr_t p = ((uintptr_t)ws + 15u) & ~(uintptr_t)15u;  // 16B align for b128
    ws = (char*)p;
    float* candScore = (float*)ws;           ws += (size_t)B * CAP * sizeof(float);
    float* candBox   = (float*)ws;

    zero_hist_kernel<<<(B * NBINS + 255) / 256, 256, 0, stream>>>(hist);

    dim3 grid(GRID_X, B);
    hist_kernel<<<grid, THREADS, 0, stream>>>(cls, hist);
    thresh_kernel<<<1, 32, 0, stream>>>(hist, tbin, candCount);
    compact_kernel<<<grid, THREADS, 0, stream>>>(cls, reg, anchors, tbin,
                                                 candCount, candScore, candBox);
    nms_kernel<<<B, THREADS, 0, stream>>>(candCount, candScore, candBox, out);
}